// KnowledgeDissemination_45603962748958
// MI455X (gfx1250) — compile-verified
//
#include <hip/hip_runtime.h>
#include <stdint.h>

// ---------------------------------------------------------------------------
// Memory-bound rotated-box-mask kernel for MI455X (gfx1250).
//   reads  ~134 MB, writes ~335 MB  ->  ~20 us floor at 23.3 TB/s.
// Strategy: analytic mask (no gather), 128-bit lanes, async-to-LDS double
// buffering (ASYNCcnt), non-temporal stores for the write-once outputs.
// ---------------------------------------------------------------------------

typedef float v4f __attribute__((ext_vector_type(4)));
typedef int   v4i __attribute__((ext_vector_type(4)));

#define AS1 __attribute__((address_space(1)))
#define AS3 __attribute__((address_space(3)))

#if __has_builtin(__builtin_amdgcn_global_load_async_to_lds_b128)
#define USE_ASYNC_LDS 1
#else
#define USE_ASYNC_LDS 0
#endif

__device__ __forceinline__ void wait_async_le1() {
#if __has_builtin(__builtin_amdgcn_s_wait_asynccnt)
    __builtin_amdgcn_s_wait_asynccnt(1);
#else
    asm volatile("s_wait_asynccnt 0x1" ::: "memory");
#endif
}
__device__ __forceinline__ void wait_async_le0() {
#if __has_builtin(__builtin_amdgcn_s_wait_asynccnt)
    __builtin_amdgcn_s_wait_asynccnt(0);
#else
    asm volatile("s_wait_asynccnt 0x0" ::: "memory");
#endif
}

// Block = 256 threads = 8 wave32. Each block owns 32 rows (1 image quarter),
// processed as 4 iterations of 8 rows (1024 floats = 4 KB) with ping-pong LDS.
__global__ void __launch_bounds__(256)
rot_mask_stream(const float* __restrict__ image,
                const float* __restrict__ azimuth,
                const float* __restrict__ alpha,
                float* __restrict__ out_target,   // B*64*64
                float* __restrict__ out_cut,      // B*64*64
                float* __restrict__ out_scene,    // B*128*128
                float* __restrict__ out_raw)      // B*128*128
{
    __shared__ float lbuf[2][1024];

    const int tid   = threadIdx.x;
    const int b     = blockIdx.x >> 2;          // image index (uniform)
    const int q     = blockIdx.x & 3;           // quarter of the image
    const int rbase = q * 32;                   // first row of this block
    const int rlane = tid >> 5;                 // row within 8-row chunk
    const int c0    = (tid & 31) * 4;           // first of 4 owned columns

    const float* src0 = image + ((size_t)b << 14) + (size_t)rbase * 128;

#if USE_ASYNC_LDS
    // Prime the pipeline: async copy chunk 0 into LDS buffer 0.
    __builtin_amdgcn_global_load_async_to_lds_b128(
        (AS1 v4i*)(src0 + (size_t)tid * 4),
        (AS3 v4i*)(&lbuf[0][tid * 4]), 0, 0);
#endif

    // ---- uniform per-batch scalars (overlapped with the async load) -------
    const float az = azimuth[b];
    float sn, cs;
    sincosf(az * (-0.017453292519943295f), &sn, &cs);   // angle = -azimuth

    const float xup   = rintf(64.0f + alpha[0] * 35.0f);  // row upper bound
    const float xdown = rintf(64.0f - alpha[1] * 35.0f);  // row lower bound
    const float yup   = rintf(64.0f + alpha[2] * 35.0f);  // col upper bound
    const float ydown = rintf(64.0f - alpha[3] * 35.0f);  // col lower bound

    const v4f one = (v4f)1.0f;

#pragma unroll
    for (int it = 0; it < 4; ++it) {
        v4f v;
#if USE_ASYNC_LDS
        if (it < 3) {
            // Prefetch next chunk into the other buffer, then wait for the
            // current one (loads complete in order; <=1 outstanding is ours).
            __builtin_amdgcn_global_load_async_to_lds_b128(
                (AS1 v4i*)(src0 + (size_t)(it + 1) * 1024 + (size_t)tid * 4),
                (AS3 v4i*)(&lbuf[(it + 1) & 1][tid * 4]), 0, 0);
            wait_async_le1();
        } else {
            wait_async_le0();
        }
        v = *(const v4f*)(&lbuf[it & 1][tid * 4]);
#else
        v = __builtin_nontemporal_load(
                (const v4f*)(src0 + (size_t)it * 1024 + (size_t)tid * 4));
#endif

        const int   r  = rbase + it * 8 + rlane;
        const float Yf = (float)r - 63.5f;
        const float sY = sn * Yf + 64.0f;   // xin + 64 partial
        const float cY = cs * Yf + 64.0f;   // yin + 64 partial

        // Analytic nearest-neighbor rotated-box mask (matches reference math).
        v4f m;
#pragma unroll
        for (int i = 0; i < 4; ++i) {
            const float X    = (float)(c0 + i) - 63.5f;
            const float colf = floorf(cs * X + sY);     // floor(xin + 64)
            const float rowf = floorf(cY - sn * X);     // floor(yin + 64)
            const bool valid = (colf >= 0.0f) && (colf < 128.0f) &&
                               (rowf >= 0.0f) && (rowf < 128.0f);
            const bool inbox = (rowf >= xdown) && (rowf < xup) &&
                               (colf >= ydown) && (colf < yup);
            m[i] = (valid && inbox) ? 1.0f : 0.0f;
        }

        const size_t pix = ((size_t)b << 14) + (size_t)r * 128 + (size_t)c0;

        // Write-once streaming outputs: non-temporal, 128-bit.
        __builtin_nontemporal_store(v,              (v4f*)(out_raw   + pix));
        __builtin_nontemporal_store(v * (one - m),  (v4f*)(out_scene + pix));

        // Center 64x64 crop (col range [32,96) is float4-aligned).
        if ((unsigned)(r - 32) < 64u && (unsigned)(c0 - 32) < 64u) {
            const size_t cpix = ((size_t)b << 12) +
                                (size_t)(r - 32) * 64 + (size_t)(c0 - 32);
            __builtin_nontemporal_store(v * m, (v4f*)(out_target + cpix));
            v4f cut = v + v;
#pragma unroll
            for (int i = 0; i < 4; ++i) cut[i] = fminf(cut[i], 1.0f);
            __builtin_nontemporal_store(cut,   (v4f*)(out_cut + cpix));
        }
    }
}

extern "C" void kernel_launch(void* const* d_in, const int* in_sizes, int n_in,
                              void* d_out, int out_size, void* d_ws, size_t ws_size,
                              hipStream_t stream) {
    const float* image   = (const float*)d_in[0];
    const float* azimuth = (const float*)d_in[1];
    const float* alpha   = (const float*)d_in[2];

    const int B = in_sizes[1];                 // 2048
    float* out        = (float*)d_out;
    float* out_target = out;                                    // B*4096
    float* out_cut    = out_target + (size_t)B * 4096;          // B*4096
    float* out_scene  = out_cut    + (size_t)B * 4096;          // B*16384
    float* out_raw    = out_scene  + (size_t)B * 16384;         // B*16384

    dim3 grid(B * 4), block(256);
    hipLaunchKernelGGL(rot_mask_stream, grid, block, 0, stream,
                       image, azimuth, alpha,
                       out_target, out_cut, out_scene, out_raw);
}